// PolyformerEmbedding_5351529250952
// MI455X (gfx1250) — compile-verified
//
#include <hip/hip_runtime.h>
#include <hip/hip_bf16.h>
#include <math.h>

typedef __attribute__((ext_vector_type(2))) float v2f;
typedef __attribute__((ext_vector_type(8))) float v8f;

#define N_BATCH    16384
#define SIG_STRIDE 8192     // 4096 * 2 floats per batch row
#define KDEG       64       // number of Chebyshev coefficients

// One wave (32 lanes) per block; each wave handles 8 batch elements.
// Rows of the 16x64 "A" matrix: row (2b+p) = normalized Chebyshev row of
// batch b, channel p.  G = A * A^T via 16 chained f32 WMMA 16x16x4 ops.
__global__ __launch_bounds__(32)
void polyformer_cheb_fit_kernel(const float* __restrict__ X,
                                float* __restrict__ out) {
    const int lane = threadIdx.x;      // 0..31
    const int m    = lane & 15;        // WMMA M / N index
    const int h    = lane >> 4;        // lane half (K-slot selector)
    const int bb   = blockIdx.x * 8;   // first batch handled by this wave

    __shared__ float Gs[16][16];       // Gram tile dump
    __shared__ float sols[8][2];       // per-batch 2x2 solution

    // ---------- Phase 1: build A rows on the fly, Gram via f32 WMMA ----------
    {
        const int  b    = m >> 1;      // batch within wave (0..7)
        const int  p    = m & 1;       // channel
        const long base = (long)(bb + b) * SIG_STRIDE;
        const float u = X[base + 126];             // x[b, 63, 0]
        const float v = X[base + 127];             // x[b, 63, 1]

        // Hoist the channel select out of the loop: Ta tracks THIS lane's
        // channel, Tb the other one.  s = Ta^2 + Tb^2 is symmetric, so the
        // per-iteration cndmask disappears entirely.
        const float xa  = p ? v : u;
        const float xb  = p ? u : v;
        const float xa2 = xa + xa;
        const float xb2 = xb + xb;

        float Ta = 1.0f, Tb = 1.0f;                // T_0
        float TaN = xa, TbN = xb;                  // T_1
        v8f acc = {};

        #pragma unroll
        for (int kb = 0; kb < KDEG; kb += 4) {
            float av[4];
            #pragma unroll
            for (int q = 0; q < 4; ++q) {
                const float s = Ta * Ta + Tb * Tb;
                av[q] = Ta * __builtin_amdgcn_rsqf(s);   // a_pk
                float t;
                t = xa2 * TaN - Ta; Ta = TaN; TaN = t;
                t = xb2 * TbN - Tb; Tb = TbN; TbN = t;
            }
            // A 16x4 layout: lanes 0-15 own K={0,1}, lanes 16-31 own K={2,3}
            // -> only 2 selects per WMMA group instead of 8.
            v2f a;
            a.x = h ? av[2] : av[0];
            a.y = h ? av[3] : av[1];
            // symmetric Gram: A- and B-operand register images coincide
            acc = __builtin_amdgcn_wmma_f32_16x16x4_f32(
                    false, a, false, a, (short)0, acc, false, false);
        }

        // C/D layout: VGPR r -> row r (lanes 0-15) / row r+8 (lanes 16-31)
        const int rowBase = h << 3;
        #pragma unroll
        for (int r = 0; r < 8; ++r)
            Gs[rowBase + r][m] = acc[r];
    }
    __syncthreads();

    // ---------- Phase 2: closed-form 2x2 solve (lanes 0..7) ----------
    if (lane < 8) {
        const int  b    = lane;
        const long base = (long)(bb + b) * SIG_STRIDE;
        const float y0  = X[base + 254];           // x[b, 127, 0]
        const float y1  = X[base + 255];           // x[b, 127, 1]
        const float G00 = Gs[2 * b][2 * b];
        const float G01 = Gs[2 * b][2 * b + 1];
        const float G11 = Gs[2 * b + 1][2 * b + 1];
        const float invd = 1.0f / (G00 * G11 - G01 * G01);  // exact: 8 lanes, once
        sols[b][0] = (G11 * y0 - G01 * y1) * invd;
        sols[b][1] = (G00 * y1 - G01 * y0) * invd;
    }
    __syncthreads();

    // ---- Phase 3: c_k = (T_k(u) s0 + T_k(v) s1) / (T_k(u)^2 + T_k(v)^2) ----
    {
        const int  b    = lane >> 2;   // 4 lanes per batch
        const int  j    = lane & 3;    // each lane owns k in [16j, 16j+16)
        const long base = (long)(bb + b) * SIG_STRIDE;
        const float u  = X[base + 126];
        const float v  = X[base + 127];
        const float u2 = u + u;
        const float v2 = v + v;
        const float s0 = sols[b][0];
        const float s1 = sols[b][1];
        float* op = out + (long)(bb + b) * KDEG;

        float Tu = 1.0f, Tv = 1.0f;
        float TuN = u, TvN = v;
        float4 buf;
        #pragma unroll
        for (int k = 0; k < KDEG; ++k) {
            const float s  = Tu * Tu + Tv * Tv;
            const float ck = (Tu * s0 + Tv * s1) * __builtin_amdgcn_rcpf(s);
            // component index is compile-time under full unroll: no selects
            const int c = k & 3;
            if      (c == 0) buf.x = ck;
            else if (c == 1) buf.y = ck;
            else if (c == 2) buf.z = ck;
            else             buf.w = ck;
            // one exec-masked 16B store per 4 coefficients, 16B-aligned
            if (c == 3 && (k >> 4) == j)
                *reinterpret_cast<float4*>(op + (k - 3)) = buf;
            float t;
            t = u2 * TuN - Tu; Tu = TuN; TuN = t;
            t = v2 * TvN - Tv; Tv = TvN; TvN = t;
        }
    }
}

extern "C" void kernel_launch(void* const* d_in, const int* in_sizes, int n_in,
                              void* d_out, int out_size, void* d_ws, size_t ws_size,
                              hipStream_t stream) {
    const float* X   = (const float*)d_in[0];
    float*       out = (float*)d_out;
    (void)in_sizes; (void)n_in; (void)out_size; (void)d_ws; (void)ws_size;

    dim3 grid(N_BATCH / 8);   // 2048 waves, 8 batch rows each
    dim3 block(32);           // one wave32 per block
    hipLaunchKernelGGL(polyformer_cheb_fit_kernel, grid, block, 0, stream, X, out);
}